// ECGEConv_8418135900673
// MI455X (gfx1250) — compile-verified
//
#include <hip/hip_runtime.h>
#include <hip/hip_bf16.h>

typedef __attribute__((ext_vector_type(2))) float v2f;
typedef __attribute__((ext_vector_type(8))) float v8f;

#define DIN  128
#define DOUT 128
#define NTYPES 4

// ---------------------------------------------------------------------------
// Utility: zero a float buffer
// ---------------------------------------------------------------------------
__global__ void zero_f32(float* __restrict__ p, long long n) {
    long long i = (long long)blockIdx.x * blockDim.x + threadIdx.x;
    if (i < n) p[i] = 0.0f;
}

// ---------------------------------------------------------------------------
// Degree: deg[col[e]] += 1
// ---------------------------------------------------------------------------
__global__ void deg_kernel(const long long* __restrict__ col,
                           float* __restrict__ deg, int E) {
    int e = blockIdx.x * blockDim.x + threadIdx.x;
    if (e < E) atomicAdd(&deg[col[e]], 1.0f);
}

// ---------------------------------------------------------------------------
// deg -> deg^{-1/2} (in place), 0 where deg == 0
// ---------------------------------------------------------------------------
__global__ void dis_kernel(float* __restrict__ deg, int N) {
    int i = blockIdx.x * blockDim.x + threadIdx.x;
    if (i < N) {
        float d = deg[i];
        deg[i] = (d > 0.0f) ? rsqrtf(d) : 0.0f;
    }
}

// ---------------------------------------------------------------------------
// norm[e] = dis[row]*dis[col]*w[e]
// ---------------------------------------------------------------------------
__global__ void norm_kernel(const long long* __restrict__ row,
                            const long long* __restrict__ col,
                            const float* __restrict__ ew,
                            const float* __restrict__ dis,
                            float* __restrict__ norm, int E) {
    int e = blockIdx.x * blockDim.x + threadIdx.x;
    if (e < E) norm[e] = dis[row[e]] * dis[col[e]] * ew[e];
}

// ---------------------------------------------------------------------------
// WMMA GEMM core: 32 K-steps of V_WMMA_F32_16X16X4_F32 over 8 N-tiles.
// B fragments are pre-packed in LDS as v2f = (W[k][n], W[k+1][n]) so each
// fragment is a single aligned ds_load_b64 into a consecutive VGPR pair.
// ---------------------------------------------------------------------------
template <bool MASKED>
__device__ __forceinline__ void gemm_core(const float* __restrict__ arow,
                                          const v2f* __restrict__ Bbase,
                                          float amask, v8f acc[8]) {
#pragma unroll 4
    for (int ks = 0; ks < DIN / 4; ++ks) {
        v2f A = *(const v2f*)(arow + 4 * ks);     // K = {4ks+klo, 4ks+klo+1}
        if (MASKED) { A.x *= amask; A.y *= amask; }
        const v2f* Bk = Bbase + 2 * ks * DOUT;    // row (k+klo)/2 of packed W
#pragma unroll
        for (int j = 0; j < 8; ++j) {
            acc[j] = __builtin_amdgcn_wmma_f32_16x16x4_f32(
                false, A, false, Bk[j * 16], (short)0, acc[j], false, false);
        }
    }
}

// ---------------------------------------------------------------------------
// WMMA GEMM: xt[t] = x @ W[t]  (fp32).
// Block: 256 threads = 8 waves; each block does 128 rows x one type.
// Each wave owns a 16x128 output strip (8 accumulator tiles of 16x16).
// ---------------------------------------------------------------------------
__global__ __launch_bounds__(256)
void rgcn_gemm_wmma(const float* __restrict__ x, const float* __restrict__ W,
                    float* __restrict__ xt, int N) {
    // Fragment-layout W: Wl[k2 * DOUT + n] = (W[2*k2][n], W[2*k2+1][n]); 64 KB
    __shared__ v2f Wl[(DIN / 2) * DOUT];

    const int t     = blockIdx.y;
    const int mbase = blockIdx.x * 128;

    // Stage W[t] into fragment layout: 2 coalesced b128 reads -> 4 b64 LDS stores
    {
        const float* Wg = W + (size_t)t * DIN * DOUT;
        for (int idx = threadIdx.x; idx < (DIN / 2) * (DOUT / 4); idx += 256) {
            const int k2 = idx >> 5;            // DOUT/4 = 32 col-groups per k2
            const int n4 = (idx & 31) * 4;
            const float4 a = *(const float4*)(Wg + (2 * k2    ) * DOUT + n4);
            const float4 b = *(const float4*)(Wg + (2 * k2 + 1) * DOUT + n4);
            v2f* dst = &Wl[k2 * DOUT + n4];
            dst[0] = v2f{a.x, b.x};
            dst[1] = v2f{a.y, b.y};
            dst[2] = v2f{a.z, b.z};
            dst[3] = v2f{a.w, b.w};
        }
    }
    __syncthreads();

    const int wave = threadIdx.x >> 5;
    const int lane = threadIdx.x & 31;
    const int l16  = lane & 15;
    const int hi   = lane >> 4;          // 0 for lanes 0-15, 1 for 16-31
    const int klo  = hi * 2;             // A/B K-offset held by this half-wave
    const int m0   = mbase + wave * 16;

    // B fragment base for this lane: packed row klo/2, column l16
    const v2f* Bbase = &Wl[(klo >> 1) * DOUT + l16];

    v8f acc[8];
#pragma unroll
    for (int j = 0; j < 8; ++j) acc[j] = (v8f)(0.0f);

    if (mbase + 128 <= N) {
        // ---- fast path: all rows valid, A loads feed WMMA directly ----
        const float* arow = x + (size_t)(m0 + l16) * DIN + klo;
        gemm_core<false>(arow, Bbase, 1.0f, acc);

#pragma unroll
        for (int j = 0; j < 8; ++j) {
            const int ncol = j * 16 + l16;
#pragma unroll
            for (int r = 0; r < 8; ++r) {
                const int rowm = m0 + r + hi * 8;
                xt[((size_t)t * N + rowm) * DOUT + ncol] = acc[j][r];
            }
        }
    } else {
        // ---- boundary path: clamp row, mask A values, guard stores ----
        int   rA    = m0 + l16;
        float amask = (rA < N) ? 1.0f : 0.0f;
        rA = (rA < N) ? rA : (N - 1);
        const float* arow = x + (size_t)rA * DIN + klo;
        gemm_core<true>(arow, Bbase, amask, acc);

#pragma unroll
        for (int j = 0; j < 8; ++j) {
            const int ncol = j * 16 + l16;
#pragma unroll
            for (int r = 0; r < 8; ++r) {
                const int rowm = m0 + r + hi * 8;
                if (rowm < N)
                    xt[((size_t)t * N + rowm) * DOUT + ncol] = acc[j][r];
            }
        }
    }
}

// ---------------------------------------------------------------------------
// Edge scatter: one wave per edge; lane handles 4 floats (float4 gather,
// 4x global_atomic_add_f32 scatter)
// ---------------------------------------------------------------------------
__global__ __launch_bounds__(256)
void edge_scatter(const long long* __restrict__ row,
                  const long long* __restrict__ col,
                  const long long* __restrict__ etype,
                  const float* __restrict__ norm,
                  const float* __restrict__ xt,
                  float* __restrict__ out, int E, int N) {
    const long long gtid = (long long)blockIdx.x * blockDim.x + threadIdx.x;
    const int e    = (int)(gtid >> 5);
    const int lane = threadIdx.x & 31;
    if (e >= E) return;

    const long long r = row[e];
    const long long c = col[e];
    const long long t = etype[e];
    const float    nm = norm[e];

    const float4 v = ((const float4*)(xt + ((size_t)t * N + c) * DOUT))[lane];
    float* dst = out + (size_t)r * DOUT + lane * 4;
    atomicAdd(dst + 0, v.x * nm);
    atomicAdd(dst + 1, v.y * nm);
    atomicAdd(dst + 2, v.z * nm);
    atomicAdd(dst + 3, v.w * nm);
}

// ---------------------------------------------------------------------------
// Finalize: out = leaky_relu(out + bias, 0.01)
// ---------------------------------------------------------------------------
__global__ void finalize_kernel(float* __restrict__ out,
                                const float* __restrict__ bias, long long n) {
    long long i = (long long)blockIdx.x * blockDim.x + threadIdx.x;
    if (i < n) {
        float v = out[i] + bias[i & (DOUT - 1)];
        out[i] = (v > 0.0f) ? v : 0.01f * v;
    }
}

// ---------------------------------------------------------------------------
// Launch
// ---------------------------------------------------------------------------
extern "C" void kernel_launch(void* const* d_in, const int* in_sizes, int n_in,
                              void* d_out, int out_size, void* d_ws, size_t ws_size,
                              hipStream_t stream) {
    const float*     x     = (const float*)d_in[0];
    const long long* eidx  = (const long long*)d_in[1];
    const long long* etype = (const long long*)d_in[2];
    const float*     ew    = (const float*)d_in[3];
    const float*     W     = (const float*)d_in[4];
    const float*     bias  = (const float*)d_in[5];

    const int N = in_sizes[0] / DIN;   // 50000
    const int E = in_sizes[3];         // 600000
    const long long* row = eidx;
    const long long* col = eidx + E;

    float* out  = (float*)d_out;
    float* deg  = (float*)d_ws;        // N floats (becomes deg^-1/2 in place)
    float* norm = deg + N;             // E floats
    float* xt   = norm + E;            // NTYPES*N*DOUT floats

    const long long nout = (long long)N * DOUT;

    // zero accumulator (d_out) and degree buffer
    zero_f32<<<(unsigned)((nout + 255) / 256), 256, 0, stream>>>(out, nout);
    zero_f32<<<(unsigned)((N + 255) / 256), 256, 0, stream>>>(deg, N);

    // degree, deg^-1/2, per-edge norm
    deg_kernel<<<(E + 255) / 256, 256, 0, stream>>>(col, deg, E);
    dis_kernel<<<(N + 255) / 256, 256, 0, stream>>>(deg, N);
    norm_kernel<<<(E + 255) / 256, 256, 0, stream>>>(row, col, ew, deg, norm, E);

    // grouped GEMM via WMMA: xt[t] = x @ W[t]
    dim3 ggrid((N + 127) / 128, NTYPES);
    rgcn_gemm_wmma<<<ggrid, 256, 0, stream>>>(x, W, xt, N);

    // gather + scatter-add (1 wave per edge)
    const long long sthreads = (long long)E * 32;
    edge_scatter<<<(unsigned)((sthreads + 255) / 256), 256, 0, stream>>>(
        row, col, etype, norm, xt, out, E, N);

    // bias + leaky relu
    finalize_kernel<<<(unsigned)((nout + 255) / 256), 256, 0, stream>>>(out, bias, nout);
}